// ProbSparseAttention_14216341750132
// MI455X (gfx1250) — compile-verified
//
#include <hip/hip_runtime.h>
#include <hip/hip_bf16.h>

// ---------------------------------------------------------------------------
// ProbSparse attention (Informer-style) for gfx1250.
// B=2, L=2048, d_model=512, H=8, Dk=64, u=40 (computed from in_sizes).
// GEMMs use V_WMMA_F32_16X16X4_F32 (f32 end-to-end: the whole problem is
// ~9.1 GFLOP over <50 MB of L2-resident data -> bandwidth bound on MI455X's
// 23.3 TB/s HBM / 192 MB L2, so low-precision WMMA buys nothing; f32 matches
// the reference numerics).  GEMM is LDS-tiled: a 256-thread block computes a
// 128x64 tile, staging A(128x64) and W(64x64) K-chunks through WGP LDS with
// bank-conflict-free padded rows (stride 68), cutting L2 traffic ~3.3x vs the
// register-direct version.  Next-chunk global_prefetch_b8 overlaps the tail.
// ---------------------------------------------------------------------------

typedef __attribute__((ext_vector_type(2))) float v2f;
typedef __attribute__((ext_vector_type(8))) float v8f;

#define NEG_INF (-__builtin_inff())

// Problem constants
#define BB 2
#define LL 2048
#define DM 512
#define HH 8
#define DK 64
#define BHL (BB*HH*LL)          // 32768
#define QKV_ELEMS (BB*HH*LL*DK) // 2097152

// GEMM tiling
#define KC 64                   // K chunk staged in LDS
#define PAD 68                  // padded row stride (floats): 68%64==4 -> no bank conflicts, 16B aligned

// ---------------------------------------------------------------------------
// C(4096x512) = A(4096x512) @ W^T(512x512) + bias, f32 WMMA 16x16x4.
// OUT_MODE 0: row-major (m,n).  OUT_MODE 1: scatter to (B,H,L,Dk) layout.
// Block = 256 threads (8 waves) -> 128x64 output tile; wave w owns rows
// [w*16, w*16+16).  Grid = (4096/128)*(512/64) = 256 blocks.
// ---------------------------------------------------------------------------
template <int OUT_MODE>
__global__ __launch_bounds__(256) void gemm_xWT(const float* __restrict__ A,
                                                const float* __restrict__ W,
                                                const float* __restrict__ bias,
                                                float* __restrict__ C) {
    __shared__ float As[128 * PAD];  // 34.8 KB
    __shared__ float Ws[64 * PAD];   // 17.4 KB

    const int tid    = threadIdx.x;
    const int lane   = tid & 31;
    const int w      = tid >> 5;              // wave in block: 0..7
    const int mBlock = blockIdx.x >> 3;       // 0..31
    const int nBlock = blockIdx.x & 7;        // 0..7
    const int mBase  = mBlock * 128;
    const int nBase  = nBlock * 64;

    const int laneM = lane & 15;
    const int kh    = (lane >> 4) << 1;       // 0 or 2 (A 16x4 / B 4x16 lane layout)

    v8f acc0 = {}, acc1 = {}, acc2 = {}, acc3 = {};

    // Cooperative-load geometry: float4 granules over a 64-float-wide K chunk.
    // A tile: 128 rows * 16 granules = 2048 -> 8 per thread.
    // W tile:  64 rows * 16 granules = 1024 -> 4 per thread.
    for (int k0 = 0; k0 < DM; k0 += KC) {
        const bool more = (k0 + KC) < DM;
#pragma unroll
        for (int i = 0; i < 8; ++i) {
            const int f   = tid + i * 256;
            const int row = f >> 4;
            const int c4  = (f & 15) << 2;
            const float* src = A + (size_t)(mBase + row) * DM + k0 + c4;
            *(float4*)&As[row * PAD + c4] = *(const float4*)src;
            if (more) __builtin_prefetch(src + KC, 0, 1);  // global_prefetch_b8
        }
#pragma unroll
        for (int i = 0; i < 4; ++i) {
            const int f   = tid + i * 256;
            const int row = f >> 4;
            const int c4  = (f & 15) << 2;
            const float* src = W + (size_t)(nBase + row) * DM + k0 + c4;
            *(float4*)&Ws[row * PAD + c4] = *(const float4*)src;
            if (more) __builtin_prefetch(src + KC, 0, 1);
        }
        __syncthreads();

        const float* __restrict__ ar = &As[(w * 16 + laneM) * PAD + kh];
        const float* __restrict__ w0 = &Ws[(laneM +  0) * PAD + kh];
        const float* __restrict__ w1 = &Ws[(laneM + 16) * PAD + kh];
        const float* __restrict__ w2 = &Ws[(laneM + 32) * PAD + kh];
        const float* __restrict__ w3 = &Ws[(laneM + 48) * PAD + kh];
#pragma unroll
        for (int k = 0; k < KC; k += 4) {
            v2f a, b0, b1, b2, b3;
            a.x  = ar[k];     a.y  = ar[k + 1];
            b0.x = w0[k];     b0.y = w0[k + 1];
            b1.x = w1[k];     b1.y = w1[k + 1];
            b2.x = w2[k];     b2.y = w2[k + 1];
            b3.x = w3[k];     b3.y = w3[k + 1];
            acc0 = __builtin_amdgcn_wmma_f32_16x16x4_f32(false, a, false, b0, (short)0, acc0, false, false);
            acc1 = __builtin_amdgcn_wmma_f32_16x16x4_f32(false, a, false, b1, (short)0, acc1, false, false);
            acc2 = __builtin_amdgcn_wmma_f32_16x16x4_f32(false, a, false, b2, (short)0, acc2, false, false);
            acc3 = __builtin_amdgcn_wmma_f32_16x16x4_f32(false, a, false, b3, (short)0, acc3, false, false);
        }
        __syncthreads();
    }

    // C/D layout: VGPR v holds row (v + 8*(lane>=16)), column laneM.
    const int rBase = mBase + w * 16 + ((lane >> 4) << 3);
    const int n0    = nBase + laneM;
    v8f accs[4] = {acc0, acc1, acc2, acc3};
#pragma unroll
    for (int j = 0; j < 4; ++j) {
        const int n  = n0 + j * 16;
        const float bn = bias[n];
#pragma unroll
        for (int v = 0; v < 8; ++v) {
            const int r = rBase + v;
            const float val = accs[j][v] + bn;
            if (OUT_MODE == 0) {
                C[(size_t)r * DM + n] = val;
            } else {
                const int b = r >> 11;       // row = b*2048 + l
                const int l = r & (LL - 1);
                const int h = n >> 6;        // col = h*64 + d
                const int d = n & (DK - 1);
                C[(((size_t)(b * HH + h) * LL) + l) * DK + d] = val;
            }
        }
    }
}

// ---------------------------------------------------------------------------
// M[b,h,q] = max_s(Q[b,h,q]·K[b,h,idx[q,s]]) - sum_s(...)/L.  One wave/query.
// ---------------------------------------------------------------------------
__global__ __launch_bounds__(256) void sample_m_kernel(const float* __restrict__ Q,
                                                       const float* __restrict__ K,
                                                       const int* __restrict__ idx,
                                                       float* __restrict__ Mv, int u) {
    const int wave = blockIdx.x * 8 + (threadIdx.x >> 5);  // 0..32767
    const int lane = threadIdx.x & 31;
    const int q    = wave & (LL - 1);
    const int bh   = wave >> 11;

    const float* __restrict__ qrow = Q + ((size_t)bh * LL + q) * DK;
    const float q0 = qrow[lane];
    const float q1 = qrow[lane + 32];
    const float* __restrict__ Kb   = K + (size_t)bh * LL * DK;
    const int*   __restrict__ irow = idx + (size_t)q * u;

    float mx = NEG_INF, sm = 0.f;
    for (int s = 0; s < u; ++s) {
        const float* kr = Kb + (size_t)irow[s] * DK;
        float p = q0 * kr[lane] + q1 * kr[lane + 32];
#pragma unroll
        for (int off = 16; off > 0; off >>= 1) p += __shfl_xor(p, off, 32);
        mx = fmaxf(mx, p);
        sm += p;
    }
    if (lane == 0) Mv[wave] = mx - sm * (1.0f / (float)LL);
}

// ---------------------------------------------------------------------------
// Top-u of M per (b,h): iterative argmax with smallest-index tie-break.
// ---------------------------------------------------------------------------
__global__ __launch_bounds__(256) void topk_kernel(float* __restrict__ Mv,
                                                   int* __restrict__ Mtop, int u) {
    const int bh = blockIdx.x;               // 0..15
    float* row = Mv + (size_t)bh * LL;
    __shared__ float svals[256];
    __shared__ int   sidx[256];

    for (int t = 0; t < u; ++t) {
        float best = NEG_INF;
        int   bidx = 0x7fffffff;
        for (int i = threadIdx.x; i < LL; i += 256) {
            const float v = row[i];
            if (v > best || (v == best && i < bidx)) { best = v; bidx = i; }
        }
        svals[threadIdx.x] = best;
        sidx[threadIdx.x]  = bidx;
        __syncthreads();
        for (int off = 128; off > 0; off >>= 1) {
            if (threadIdx.x < off) {
                const float ov = svals[threadIdx.x + off];
                const int   oi = sidx[threadIdx.x + off];
                if (ov > svals[threadIdx.x] ||
                    (ov == svals[threadIdx.x] && oi < sidx[threadIdx.x])) {
                    svals[threadIdx.x] = ov;
                    sidx[threadIdx.x]  = oi;
                }
            }
            __syncthreads();
        }
        if (threadIdx.x == 0) {
            Mtop[bh * u + t] = sidx[0];
            row[sidx[0]]     = NEG_INF;
        }
        __syncthreads();
    }
}

// ---------------------------------------------------------------------------
// ctx[bh,t] = softmax(Q[bh,Mtop[t]]·K^T * 1/8) @ V.  One block per (bh,t).
// ---------------------------------------------------------------------------
__global__ __launch_bounds__(256) void sparse_attn_kernel(const float* __restrict__ Q,
                                                          const float* __restrict__ K,
                                                          const float* __restrict__ V,
                                                          const int* __restrict__ Mtop,
                                                          float* __restrict__ ctx, int u) {
    const int t  = blockIdx.x % u;
    const int bh = blockIdx.x / u;

    __shared__ float qs[DK];
    __shared__ float sc[LL];
    __shared__ float red[256];
    __shared__ float redc[4 * DK];

    const int l = Mtop[bh * u + t];
    const float* __restrict__ Kb = K + (size_t)bh * LL * DK;
    const float* __restrict__ Vb = V + (size_t)bh * LL * DK;

    if (threadIdx.x < DK)
        qs[threadIdx.x] = Q[((size_t)bh * LL + l) * DK + threadIdx.x];
    __syncthreads();

    const float scale = 0.125f;  // 1/sqrt(64)
    for (int k = threadIdx.x; k < LL; k += 256) {
        const float* kr = Kb + (size_t)k * DK;
        float d = 0.f;
#pragma unroll 8
        for (int e = 0; e < DK; ++e) d += qs[e] * kr[e];
        sc[k] = d * scale;
    }
    __syncthreads();

    // block max
    float lm = NEG_INF;
    for (int k = threadIdx.x; k < LL; k += 256) lm = fmaxf(lm, sc[k]);
    red[threadIdx.x] = lm;
    __syncthreads();
    for (int off = 128; off > 0; off >>= 1) {
        if (threadIdx.x < off)
            red[threadIdx.x] = fmaxf(red[threadIdx.x], red[threadIdx.x + off]);
        __syncthreads();
    }
    const float mx = red[0];
    __syncthreads();

    // exp + block sum
    float ls = 0.f;
    for (int k = threadIdx.x; k < LL; k += 256) {
        const float e = __expf(sc[k] - mx);
        sc[k] = e;
        ls += e;
    }
    red[threadIdx.x] = ls;
    __syncthreads();
    for (int off = 128; off > 0; off >>= 1) {
        if (threadIdx.x < off) red[threadIdx.x] += red[threadIdx.x + off];
        __syncthreads();
    }
    const float tot = red[0];

    // attn @ V : 4 key-range groups x 64 dims
    const int d = threadIdx.x & (DK - 1);
    const int g = threadIdx.x >> 6;
    float acc = 0.f;
    for (int k = g * 512; k < (g + 1) * 512; ++k)
        acc += sc[k] * Vb[(size_t)k * DK + d];
    redc[g * DK + d] = acc;
    __syncthreads();
    if (g == 0) {
        const float s = redc[d] + redc[DK + d] + redc[2 * DK + d] + redc[3 * DK + d];
        ctx[((size_t)(bh * u + t)) * DK + d] = s / tot;
    }
}

// ---------------------------------------------------------------------------
// V_mean over L per (b,h,d).  1024 values total.
// ---------------------------------------------------------------------------
__global__ __launch_bounds__(256) void vmean_kernel(const float* __restrict__ V,
                                                    float* __restrict__ vm) {
    const int i  = blockIdx.x * 256 + threadIdx.x;  // 0..1023
    const int d  = i & (DK - 1);
    const int bh = i >> 6;
    const float* __restrict__ Vb = V + (size_t)bh * LL * DK + d;
    float s = 0.f;
    for (int l = 0; l < LL; ++l) s += Vb[(size_t)l * DK];
    vm[i] = s * (1.0f / (float)LL);
}

// out_pre[b,l,h*64+d] = vm[b,h,d]
__global__ __launch_bounds__(256) void fill_mean_kernel(const float* __restrict__ vm,
                                                        float* __restrict__ outp) {
    const int i = blockIdx.x * 256 + threadIdx.x;  // 0..2097151
    const int n = i & (DM - 1);
    const int m = i >> 9;
    const int b = m >> 11;
    const int h = n >> 6;
    const int d = n & (DK - 1);
    outp[i] = vm[((b << 3) + h) * DK + d];
}

// out_pre rows at Mtop <- context
__global__ __launch_bounds__(256) void scatter_ctx_kernel(const float* __restrict__ ctx,
                                                          const int* __restrict__ Mtop,
                                                          float* __restrict__ outp,
                                                          int u, int total) {
    const int i = blockIdx.x * 256 + threadIdx.x;
    if (i >= total) return;
    const int d  = i & (DK - 1);
    const int t  = (i >> 6) % u;
    const int bh = (i >> 6) / u;
    const int b  = bh >> 3;
    const int h  = bh & 7;
    const int l  = Mtop[bh * u + t];
    outp[((size_t)(b * LL + l)) * DM + h * DK + d] = ctx[i];
}

// ---------------------------------------------------------------------------
extern "C" void kernel_launch(void* const* d_in, const int* in_sizes, int n_in,
                              void* d_out, int out_size, void* d_ws, size_t ws_size,
                              hipStream_t stream) {
    const float* x  = (const float*)d_in[0];
    const float* Wq = (const float*)d_in[1];
    const float* bq = (const float*)d_in[2];
    const float* Wk = (const float*)d_in[3];
    const float* bk = (const float*)d_in[4];
    const float* Wv = (const float*)d_in[5];
    const float* bv = (const float*)d_in[6];
    const float* Wo = (const float*)d_in[7];
    const float* bo = (const float*)d_in[8];
    const int*  idx = (const int*)d_in[9];

    const int u = in_sizes[9] / LL;  // 40 for L=2048, FACTOR=5

    // Workspace layout (floats): Q | K | V | out_pre | M | ctx | vmean | Mtop
    float* ws   = (float*)d_ws;
    float* Q    = ws;
    float* Kp   = Q    + QKV_ELEMS;
    float* Vp   = Kp   + QKV_ELEMS;
    float* outp = Vp   + QKV_ELEMS;
    float* Mv   = outp + QKV_ELEMS;
    float* ctx  = Mv   + BHL;
    float* vm   = ctx  + (size_t)BB * HH * u * DK;
    int*   Mtop = (int*)(vm + BB * HH * DK);

    const dim3 gB(256), gG(256);  // 256 blocks: 32 M-blocks x 8 N-blocks

    // Q/K/V projections -> (B,H,L,Dk)
    gemm_xWT<1><<<gG, gB, 0, stream>>>(x, Wq, bq, Q);
    gemm_xWT<1><<<gG, gB, 0, stream>>>(x, Wk, bk, Kp);
    gemm_xWT<1><<<gG, gB, 0, stream>>>(x, Wv, bv, Vp);

    // Sparsity measurement + top-u selection
    sample_m_kernel<<<BHL / 8, 256, 0, stream>>>(Q, Kp, idx, Mv, u);
    topk_kernel<<<BB * HH, 256, 0, stream>>>(Mv, Mtop, u);

    // Attention on selected queries
    sparse_attn_kernel<<<BB * HH * u, 256, 0, stream>>>(Q, Kp, Vp, Mtop, ctx, u);

    // Assemble pre-projection output: mean(V) everywhere, context at Mtop rows
    vmean_kernel<<<(BB * HH * DK) / 256, 256, 0, stream>>>(Vp, vm);
    fill_mean_kernel<<<QKV_ELEMS / 256, 256, 0, stream>>>(vm, outp);
    const int scat_total = BB * HH * u * DK;
    scatter_ctx_kernel<<<(scat_total + 255) / 256, 256, 0, stream>>>(ctx, Mtop, outp, u, scat_total);

    // Final projection -> d_out (B, L, d_model) row-major
    gemm_xWT<0><<<gG, gB, 0, stream>>>(outp, Wo, bo, (float*)d_out);
}